// WeightQuantizer_9732395892920
// MI455X (gfx1250) — compile-verified
//
#include <hip/hip_runtime.h>

typedef float v2f __attribute__((ext_vector_type(2)));
typedef float v8f __attribute__((ext_vector_type(8)));
typedef float f4  __attribute__((ext_vector_type(4)));

#define NBIT        4
#define NLEVELS     16          // 2^NBIT
#define FILT_PER_WG 16          // one 16x16x4 WMMA covers 16 filters
#define WS_STRIDE   32          // per-filter: 16 levels + 15 thrs + 1 pad
#define QBLOCK      256         // threads per quantize block

// ---------------------------------------------------------------------------
// Kernel A: levels = basis @ lm^T via V_WMMA_F32_16X16X4_F32, then per-filter
// bitonic sort of the 16 levels + midpoint thresholds -> workspace.
// One wave handles 16 filters.  lm[n][k] = ((n>>k)&1)*2 - 1.
// ---------------------------------------------------------------------------
__global__ __launch_bounds__(32) void levels_kernel(const float* __restrict__ basis,
                                                    float* __restrict__ ws) {
    const int lane  = threadIdx.x;        // 0..31, full wave (EXEC all ones)
    const int m     = lane & 15;          // row (A) / column (B) index
    const int half  = lane >> 4;          // lane half selects K pair
    const int fbase = blockIdx.x * FILT_PER_WG;

    // A (16x4 f32, MxK): VGPR j, lane-half h -> K = j + 2h ; lane m -> M = m
    v2f a;
    a.x = basis[(fbase + m) * NBIT + (0 + 2 * half)];
    a.y = basis[(fbase + m) * NBIT + (1 + 2 * half)];

    // B (4x16 f32, KxN): same K striping, lane m -> N = m.  B[k][n] = lm[n][k]
    v2f b;
    b.x = (float)(((m >> (0 + 2 * half)) & 1) * 2 - 1);
    b.y = (float)(((m >> (1 + 2 * half)) & 1) * 2 - 1);

    v8f c = {};
    // 8 args: (neg_a, A, neg_b, B, c_mod, C, reuse_a, reuse_b)
    c = __builtin_amdgcn_wmma_f32_16x16x4_f32(false, a, false, b,
                                              (short)0, c, false, false);

    // C layout: VGPR r, lanes 0-15 -> (M=r, N=lane); lanes 16-31 -> (M=8+r, N=lane-16)
    __shared__ float lev[16][17];         // 17-stride dodges bank conflicts
#pragma unroll
    for (int r = 0; r < 8; ++r)
        lev[r + 8 * half][m] = c[r];
    __syncthreads();

    if (lane < FILT_PER_WG) {
        float v[NLEVELS];
#pragma unroll
        for (int i = 0; i < NLEVELS; ++i) v[i] = lev[lane][i];

        // fully-unrolled 16-element bitonic sorting network (registers only)
#pragma unroll
        for (int k = 2; k <= NLEVELS; k <<= 1) {
#pragma unroll
            for (int j = k >> 1; j > 0; j >>= 1) {
#pragma unroll
                for (int i = 0; i < NLEVELS; ++i) {
                    const int ij = i ^ j;
                    if (ij > i) {
                        const bool up = ((i & k) == 0);
                        const float lo = fminf(v[i], v[ij]);
                        const float hi = fmaxf(v[i], v[ij]);
                        v[i]  = up ? lo : hi;
                        v[ij] = up ? hi : lo;
                    }
                }
            }
        }

        float* o = ws + (size_t)(fbase + lane) * WS_STRIDE;
#pragma unroll
        for (int i = 0; i < NLEVELS; ++i) o[i] = v[i];
#pragma unroll
        for (int i = 0; i < NLEVELS - 1; ++i)
            o[NLEVELS + i] = 0.5f * (v[i] + v[i + 1]);
    }
}

// ---------------------------------------------------------------------------
// Kernel B: streaming quantize.  One block per filter; levels/thrs staged via
// LDS into registers; float4 non-temporal load -> 15-step select chain
// (idx = #thrs <= v, side='right') -> float4 non-temporal store.
// ---------------------------------------------------------------------------
__global__ __launch_bounds__(QBLOCK) void quant_kernel(const float* __restrict__ x,
                                                       const float* __restrict__ ws,
                                                       float* __restrict__ out,
                                                       int kelems) {
    const int f = blockIdx.x;

    __shared__ float sP[WS_STRIDE];
    if (threadIdx.x < WS_STRIDE - 1)      // 31 floats: 16 levels + 15 thrs
        sP[threadIdx.x] = ws[(size_t)f * WS_STRIDE + threadIdx.x];
    __syncthreads();

    float L[NLEVELS], T[NLEVELS - 1];
#pragma unroll
    for (int i = 0; i < NLEVELS; ++i)     L[i] = sP[i];
#pragma unroll
    for (int i = 0; i < NLEVELS - 1; ++i) T[i] = sP[NLEVELS + i];

    const size_t fbase = (size_t)f * kelems;
    const int per_iter = QBLOCK * 4;
    const int niter    = kelems / per_iter;   // 18432/1024 = 18

#pragma unroll 2
    for (int it = 0; it < niter; ++it) {
        const size_t off = fbase + (size_t)it * per_iter + (size_t)threadIdx.x * 4;
        f4 xv = __builtin_nontemporal_load((const f4*)(x + off));
        f4 yv;
#pragma unroll
        for (int e = 0; e < 4; ++e) {
            const float v = xv[e];
            float y = L[0];
#pragma unroll
            for (int i = 0; i < NLEVELS - 1; ++i)
                y = (v >= T[i]) ? L[i + 1] : y;   // searchsorted(..., 'right')
            yv[e] = y;
        }
        __builtin_nontemporal_store(yv, (f4*)(out + off));
    }
}

// ---------------------------------------------------------------------------
extern "C" void kernel_launch(void* const* d_in, const int* in_sizes, int n_in,
                              void* d_out, int out_size, void* d_ws, size_t ws_size,
                              hipStream_t stream) {
    const float* x     = (const float*)d_in[0];   // [F, IN_CH, 3, 3] f32
    const float* basis = (const float*)d_in[1];   // [F, 4] f32
    float* out = (float*)d_out;
    float* ws  = (float*)d_ws;                    // needs F*32*4 = 256 KB

    const int F      = in_sizes[1] / NBIT;        // 2048
    const int kelems = in_sizes[0] / F;           // 18432

    levels_kernel<<<F / FILT_PER_WG, 32, 0, stream>>>(basis, ws);
    quant_kernel<<<F, QBLOCK, 0, stream>>>(x, ws, out, kelems);
}